// MultiLoRALayerMaskingHom_44933947850966
// MI455X (gfx1250) — compile-verified
//
#include <hip/hip_runtime.h>

typedef __attribute__((ext_vector_type(2))) float v2f;
typedef __attribute__((ext_vector_type(4))) float v4f;
typedef __attribute__((ext_vector_type(8))) float v8f;

#define NTOK   1024
#define D_IN   2048
#define D_OUT  2048
#define RANK   16
#define KCHUNK 256            // stage-1 i-chunk per wave (8 chunks)
#define OCHUNK 256            // stage-2 o-chunk per wave (8 chunks)
// SCALING = ALPHA/RANK = 16/16 = 1.0f -> omitted from the math

// ---------------------------------------------------------------------------
// Kernel 1: partial P tiles.  wave w -> (tile = w>>3, chunk = w&7)
//   tile = b*64 + (n0/16);  P_part[w][16][16] row-major in ws.
// WMMA: D(16n x 16r) += X(16n x 4i) * B'(4i x 16r), B'[i][r] = B[a][r][i]
// ---------------------------------------------------------------------------
__global__ __launch_bounds__(256) void lora_stage1(
    const float* __restrict__ x, const float* __restrict__ Bm,
    const int* __restrict__ ids, float* __restrict__ ppart)
{
  const int lane = threadIdx.x & 31;
  const int wv   = threadIdx.x >> 5;
  const int lo   = lane & 15;
  const int hi   = lane >> 4;

  const int w     = blockIdx.x * 8 + wv;       // 0..4095
  const int tile  = w >> 3;                    // 0..511
  const int chunk = w & 7;                     // 0..7
  const int b     = tile >> 6;
  const int n0    = (tile & 63) << 4;
  const int a     = ids[b];

  const float* xrow = x  + ((size_t)b * NTOK + n0 + lo) * D_IN;
  const float* brow = Bm + ((size_t)a * RANK + lo) * D_IN;
  const int    off4 = hi * 4;                  // lanes 0-15: cols i..i+3, 16-31: i+4..i+7
  const int    i0   = chunk * KCHUNK;

  v8f acc = {};
  #pragma unroll 4
  for (int i = i0; i < i0 + KCHUNK; i += 8) {
    v4f xv = *(const v4f*)(xrow + i + off4);
    v4f bv = *(const v4f*)(brow + i + off4);
    v2f a0 = {xv.x, xv.y}, a1 = {xv.z, xv.w};
    v2f b0 = {bv.x, bv.y}, b1 = {bv.z, bv.w};
    acc = __builtin_amdgcn_wmma_f32_16x16x4_f32(false, a0, false, b0,
                                                (short)0, acc, false, false);
    acc = __builtin_amdgcn_wmma_f32_16x16x4_f32(false, a1, false, b1,
                                                (short)0, acc, false, false);
  }

  // D layout: vgpr j -> row (j + 8*hi), col lo.  Store P_part row-major.
  float* pb = ppart + (size_t)w * 256;
  const int mb = hi * 8;
  #pragma unroll
  for (int j = 0; j < 8; ++j)
    pb[(mb + j) * 16 + lo] = acc[j];
}

// ---------------------------------------------------------------------------
// Kernel 2: out tile.  wave w -> (tile = w>>3, ochunk = w&7)
// Reduce the 8 P partials (deterministic order) directly into the WMMA
// A-operand layout, then sweep 16 output 16x16 tiles.
// WMMA: D(16n x 16o) += P(16n x 4r) * A'(4r x 16o), A'[r][o] = A[a][o][r]
// ---------------------------------------------------------------------------
__global__ __launch_bounds__(256) void lora_stage2(
    const float* __restrict__ Am, const int* __restrict__ ids,
    const float* __restrict__ ppart, float* __restrict__ out)
{
  const int lane = threadIdx.x & 31;
  const int wv   = threadIdx.x >> 5;
  const int lo   = lane & 15;
  const int hi   = lane >> 4;
  const int koff = hi * 2;

  const int w    = blockIdx.x * 8 + wv;        // 0..4095
  const int tile = w >> 3;                     // 0..511
  const int oc   = w & 7;                      // 0..7
  const int b    = tile >> 6;
  const int n0   = (tile & 63) << 4;
  const int a    = ids[b];

  // A-operand of stage 2: lane L needs P[row lo][r = k0 + koff .. +1]
  v2f p0 = {}, p1 = {}, p2 = {}, p3 = {};
  const float* pbase = ppart + (size_t)tile * 8 * 256 + lo * 16 + koff;
  #pragma unroll
  for (int c = 0; c < 8; ++c) {
    const float* pc = pbase + c * 256;
    p0 += *(const v2f*)(pc + 0);
    p1 += *(const v2f*)(pc + 4);
    p2 += *(const v2f*)(pc + 8);
    p3 += *(const v2f*)(pc + 12);
  }

  const float* abase = Am + (size_t)a * D_OUT * RANK + (size_t)lo * RANK + koff;
  float* obase = out + ((size_t)b * NTOK + n0) * D_OUT + lo;
  const int mb = hi * 8;

  const int o0beg = oc * OCHUNK;
  for (int o0 = o0beg; o0 < o0beg + OCHUNK; o0 += 16) {
    // B-operand: lane L -> A[a][o0+lo][k0 + koff .. +1] (contiguous in r)
    const float* ap = abase + (size_t)o0 * RANK;
    v2f q0 = *(const v2f*)(ap + 0);
    v2f q1 = *(const v2f*)(ap + 4);
    v2f q2 = *(const v2f*)(ap + 8);
    v2f q3 = *(const v2f*)(ap + 12);

    v8f oacc = {};
    oacc = __builtin_amdgcn_wmma_f32_16x16x4_f32(false, p0, false, q0,
                                                 (short)0, oacc, false, false);
    oacc = __builtin_amdgcn_wmma_f32_16x16x4_f32(false, p1, false, q1,
                                                 (short)0, oacc, false, false);
    oacc = __builtin_amdgcn_wmma_f32_16x16x4_f32(false, p2, false, q2,
                                                 (short)0, oacc, false, false);
    oacc = __builtin_amdgcn_wmma_f32_16x16x4_f32(false, p3, false, q3,
                                                 (short)0, oacc, false, false);

    // D layout: vgpr j -> row (n0 + mb + j), col (o0 + lo).  SCALING == 1.0
    #pragma unroll
    for (int j = 0; j < 8; ++j)
      obase[(size_t)(mb + j) * D_OUT + o0] = oacc[j];
  }
}

extern "C" void kernel_launch(void* const* d_in, const int* in_sizes, int n_in,
                              void* d_out, int out_size, void* d_ws, size_t ws_size,
                              hipStream_t stream) {
  const float* x   = (const float*)d_in[0];   // (8,1024,2048) f32
  const float* A   = (const float*)d_in[1];   // (8,2048,16)  f32
  const float* B   = (const float*)d_in[2];   // (8,16,2048)  f32
  const int*   ids = (const int*)d_in[3];     // (8,) i32
  float* out = (float*)d_out;                 // (8,1024,2048) f32
  float* ppart = (float*)d_ws;                // 512 tiles * 8 chunks * 256 f32 = 4 MB

  // 4096 waves each: 512 blocks x 256 threads (8 waves/block)
  lora_stage1<<<dim3(512), dim3(256), 0, stream>>>(x, B, ids, ppart);
  lora_stage2<<<dim3(512), dim3(256), 0, stream>>>(A, ids, ppart, out);
}